// ClasswiseDBMInnerProduct_67765993997126
// MI455X (gfx1250) — compile-verified
//
#include <hip/hip_runtime.h>
#include <hip/hip_bf16.h>

#define BB 4096
#define DD 128
#define CC 30000

#define SCALE_F   30.0f
#define RAD2DEG_F 57.29577951308232f
#define DEG2RAD_F 0.017453292519943295f

typedef __attribute__((ext_vector_type(2))) float v2f;
typedef __attribute__((ext_vector_type(8))) float v8f;

__device__ __forceinline__ float waveReduceSum(float s) {
#pragma unroll
  for (int off = 16; off > 0; off >>= 1) s += __shfl_xor(s, off, 32);
  return s;
}

// ---------------------------------------------------------------------------
// ws scalar slots (float/uint aliased):
// [0]=sum_xnorm [1]=sum_wnorm [2]=sum_theta [3]=min_bits [4]=max_bits
// [5]=avg_theta [6]=margin_above
// ---------------------------------------------------------------------------
__global__ void init_kernel(float* scal) {
  if (threadIdx.x == 0) {
    unsigned* u = (unsigned*)scal;
    scal[0] = 0.0f; scal[1] = 0.0f; scal[2] = 0.0f;
    u[3] = 0x7F800000u;   // +inf  (min accumulator; thetas are >= 0)
    u[4] = 0x00000000u;   // 0     (max accumulator)
    scal[5] = 0.0f; scal[6] = 0.0f;
  }
}

// One wave per row: L2 normalize a row of 128 f32, atomic-add its norm.
__global__ void rownorm_kernel(const float* __restrict__ X, float* __restrict__ Xn,
                               float* __restrict__ sumAcc, int rows) {
  int wave = blockIdx.x * (blockDim.x >> 5) + (threadIdx.x >> 5);
  int lane = threadIdx.x & 31;
  if (wave >= rows) return;
  const float4 v = ((const float4*)(X + (size_t)wave * DD))[lane];
  float s = v.x * v.x + v.y * v.y + v.z * v.z + v.w * v.w;
  s = waveReduceSum(s);
  float norm = sqrtf(s);
  float inv = 1.0f / norm;
  float4 o; o.x = v.x * inv; o.y = v.y * inv; o.z = v.z * inv; o.w = v.w * inv;
  ((float4*)(Xn + (size_t)wave * DD))[lane] = o;
  if (lane == 0) atomicAdd(sumAcc, norm);
}

// One wave per sample: cos_l = <nf_b, nw_label>, theta (deg), global reductions.
__global__ void theta_kernel(const float* __restrict__ nf, const float* __restrict__ nw,
                             const int* __restrict__ label,
                             float* __restrict__ thetas, float* __restrict__ cosl,
                             float* __restrict__ scal) {
  int wave = blockIdx.x * (blockDim.x >> 5) + (threadIdx.x >> 5);
  int lane = threadIdx.x & 31;
  if (wave >= BB) return;
  int lbl = label[wave];
  const float4 a = ((const float4*)(nf + (size_t)wave * DD))[lane];
  const float4 w = ((const float4*)(nw + (size_t)lbl * DD))[lane];
  float s = a.x * w.x + a.y * w.y + a.z * w.z + a.w * w.w;
  s = waveReduceSum(s);
  if (lane == 0) {
    float cl = s;
    float c = fminf(fmaxf(cl, -1.0f), 1.0f);
    float th = acosf(c) * RAD2DEG_F;
    thetas[wave] = th;
    cosl[wave] = cl;
    atomicAdd(&scal[2], th);
    atomicMin(((unsigned*)scal) + 3, __float_as_uint(th));
    atomicMax(((unsigned*)scal) + 4, __float_as_uint(th));
  }
}

// Single block: stddev, scalar outputs, margin_above for fix-up kernel.
__global__ void stats_kernel(const float* __restrict__ thetas, float* __restrict__ scal,
                             float* __restrict__ outS) {
  __shared__ float red[256];
  float avg = scal[2] * (1.0f / (float)BB);
  float part = 0.0f;
  for (int i = threadIdx.x; i < BB; i += 256) {
    float d = thetas[i] - avg;
    part += d * d;
  }
  red[threadIdx.x] = part;
  __syncthreads();
  for (int off = 128; off > 0; off >>= 1) {
    if ((int)threadIdx.x < off) red[threadIdx.x] += red[threadIdx.x + off];
    __syncthreads();
  }
  if (threadIdx.x == 0) {
    float stdv = sqrtf(red[0] / (float)(BB - 1));
    float mn = __uint_as_float(((unsigned*)scal)[3]);
    float mx = __uint_as_float(((unsigned*)scal)[4]);
    outS[0] = avg;
    outS[1] = mn;
    outS[2] = mx;
    outS[3] = stdv;
    outS[4] = scal[1] * (1.0f / (float)CC);  // avg_w_norm
    outS[5] = scal[0] * (1.0f / (float)BB);  // avg_x_norm
    scal[5] = avg;
    scal[6] = (mx < 90.0f) ? (90.0f - avg) * DEG2RAD_F : 0.0f;
  }
}

// ---------------------------------------------------------------------------
// WMMA GEMM: cos = nf[4096x128] @ nw^T[128x30000] via v_wmma_f32_16x16x4_f32.
// Wave tile: 32(M) x 80(N): 2 A-frags x 5 B-frags -> 10 accumulators.
// 30000 % 80 == 0 and 4096 % 32 == 0 -> completely uniform control flow,
// EXEC stays all-ones (WMMA requirement), no exec-mask save/restore.
// K loop is double-buffered: fragments for step k+4 are issued before the
// WMMAs consuming step k, so loads overlap matrix math.
// ---------------------------------------------------------------------------
#define WMMA_F32(A, B, C) \
  __builtin_amdgcn_wmma_f32_16x16x4_f32(false, (A), false, (B), (short)0, (C), false, false)

__global__ void __launch_bounds__(256) gemm_kernel(const float* __restrict__ nf,
                                                   const float* __restrict__ nw,
                                                   float* __restrict__ cosOut,
                                                   float* __restrict__ margOut) {
  const int nWaves = CC / 80;  // 375
  int wave = blockIdx.x * (blockDim.x >> 5) + (threadIdx.x >> 5);
  int lane = threadIdx.x & 31;
  int mTile = wave / nWaves;           // 0..127
  int nWave = wave - mTile * nWaves;   // 0..374
  int row0 = mTile * 32;
  int col0 = nWave * 80;

  int l16 = lane & 15;
  int khalf = lane >> 4;  // 0: K pair {0,1}; 1: K pair {2,3} within each k-step

  // A fragment: lane l16 holds row (row0[+16]+l16), elements K=k+2*khalf,+1.
  const float* aP0 = nf + (size_t)(row0 + l16) * DD + 2 * khalf;
  const float* aP1 = aP0 + (size_t)16 * DD;
  // B fragment: lane l16 holds row (col0+j*16+l16) of nw (column of nw^T).
  const float* bP  = nw + (size_t)(col0 + l16) * DD + 2 * khalf;

  v8f acc00 = {}, acc01 = {}, acc02 = {}, acc03 = {}, acc04 = {};
  v8f acc10 = {}, acc11 = {}, acc12 = {}, acc13 = {}, acc14 = {};

  // Preload K-step 0.
  v2f a0 = *(const v2f*)(aP0);
  v2f a1 = *(const v2f*)(aP1);
  v2f b0 = *(const v2f*)(bP);
  v2f b1 = *(const v2f*)(bP + 16 * DD);
  v2f b2 = *(const v2f*)(bP + 32 * DD);
  v2f b3 = *(const v2f*)(bP + 48 * DD);
  v2f b4 = *(const v2f*)(bP + 64 * DD);

#pragma unroll 2
  for (int k = 0; k < DD - 4; k += 4) {
    // Prefetch K-step k+4 (issued before the WMMAs that consume step k).
    v2f na0 = *(const v2f*)(aP0 + k + 4);
    v2f na1 = *(const v2f*)(aP1 + k + 4);
    v2f nb0 = *(const v2f*)(bP + k + 4);
    v2f nb1 = *(const v2f*)(bP + 16 * DD + k + 4);
    v2f nb2 = *(const v2f*)(bP + 32 * DD + k + 4);
    v2f nb3 = *(const v2f*)(bP + 48 * DD + k + 4);
    v2f nb4 = *(const v2f*)(bP + 64 * DD + k + 4);

    acc00 = WMMA_F32(a0, b0, acc00);
    acc10 = WMMA_F32(a1, b0, acc10);
    acc01 = WMMA_F32(a0, b1, acc01);
    acc11 = WMMA_F32(a1, b1, acc11);
    acc02 = WMMA_F32(a0, b2, acc02);
    acc12 = WMMA_F32(a1, b2, acc12);
    acc03 = WMMA_F32(a0, b3, acc03);
    acc13 = WMMA_F32(a1, b3, acc13);
    acc04 = WMMA_F32(a0, b4, acc04);
    acc14 = WMMA_F32(a1, b4, acc14);

    a0 = na0; a1 = na1;
    b0 = nb0; b1 = nb1; b2 = nb2; b3 = nb3; b4 = nb4;
  }
  // Tail: consume the final buffered K-step (k = 124).
  acc00 = WMMA_F32(a0, b0, acc00);
  acc10 = WMMA_F32(a1, b0, acc10);
  acc01 = WMMA_F32(a0, b1, acc01);
  acc11 = WMMA_F32(a1, b1, acc11);
  acc02 = WMMA_F32(a0, b2, acc02);
  acc12 = WMMA_F32(a1, b2, acc12);
  acc03 = WMMA_F32(a0, b3, acc03);
  acc13 = WMMA_F32(a1, b3, acc13);
  acc04 = WMMA_F32(a0, b4, acc04);
  acc14 = WMMA_F32(a1, b4, acc14);

  // C/D layout: VGPR v -> M = v (lanes 0-15) / v+8 (lanes 16-31); N = lane&15.
  // Streamed (non-temporal) stores: outputs are written once, never re-read.
  v8f accs0[5] = {acc00, acc01, acc02, acc03, acc04};
  v8f accs1[5] = {acc10, acc11, acc12, acc13, acc14};
#pragma unroll
  for (int m = 0; m < 2; ++m) {
    size_t base = (size_t)(row0 + m * 16 + 8 * khalf) * CC + (size_t)col0 + l16;
#pragma unroll
    for (int j = 0; j < 5; ++j) {
      v8f acc = (m == 0) ? accs0[j] : accs1[j];
#pragma unroll
      for (int v = 0; v < 8; ++v) {
        size_t idx = base + (size_t)v * CC + j * 16;
        float c = acc[v];
        __builtin_nontemporal_store(c, cosOut + idx);
        __builtin_nontemporal_store(SCALE_F * fminf(fmaxf(c, -1.0f), 1.0f), margOut + idx);
      }
    }
  }
}

// Overwrite marginal_logits at [b, label[b]] with the full margin expression.
__global__ void fixup_kernel(const float* __restrict__ thetas, const float* __restrict__ cosl,
                             const int* __restrict__ label, const float* __restrict__ scal,
                             float* __restrict__ margOut) {
  int b = blockIdx.x * blockDim.x + threadIdx.x;
  if (b >= BB) return;
  float th  = thetas[b];
  float avg = scal[5];
  float mA  = scal[6];
  float m   = (th > avg) ? mA : 0.0f;  // margin_below is identically 0
  float cl  = cosl[b];
  bool cond = cl > -cosf(m);
  float add_m = cond ? m : 0.0f;
  float ext_m = cond ? 0.0f : (-m * sinf(m));
  float c = fminf(fmaxf(cl, -1.0f), 1.0f);
  margOut[(size_t)b * CC + label[b]] = SCALE_F * (cosf(acosf(c) + add_m) + ext_m);
}

extern "C" void kernel_launch(void* const* d_in, const int* in_sizes, int n_in,
                              void* d_out, int out_size, void* d_ws, size_t ws_size,
                              hipStream_t stream) {
  const float* feat    = (const float*)d_in[0];
  const int*   label   = (const int*)d_in[1];
  const float* weights = (const float*)d_in[2];

  float* out0 = (float*)d_out;                      // cos          [B*C]
  float* out1 = out0 + (size_t)BB * CC;             // marginal     [B*C]
  float* outS = out1 + (size_t)BB * CC;             // 6 scalars

  float* nf     = (float*)d_ws;                     // B*D
  float* nw     = nf + (size_t)BB * DD;             // C*D
  float* thetas = nw + (size_t)CC * DD;             // B
  float* cosl   = thetas + BB;                      // B
  float* scal   = cosl + BB;                        // 8 slots

  init_kernel<<<1, 32, 0, stream>>>(scal);

  // 8 waves (rows) per 256-thread block
  rownorm_kernel<<<BB / 8, 256, 0, stream>>>(feat, nf, &scal[0], BB);
  rownorm_kernel<<<(CC + 7) / 8, 256, 0, stream>>>(weights, nw, &scal[1], CC);

  theta_kernel<<<BB / 8, 256, 0, stream>>>(nf, nw, label, thetas, cosl, scal);
  stats_kernel<<<1, 256, 0, stream>>>(thetas, scal, outS);

  const int mWaves = BB / 32;   // 128
  const int nWaves = CC / 80;   // 375
  const int totalWaves = mWaves * nWaves;  // 48000
  gemm_kernel<<<totalWaves / 8, 256, 0, stream>>>(nf, nw, out0, out1);

  fixup_kernel<<<BB / 256, 256, 0, stream>>>(thetas, cosl, label, scal, out1);
}